// SpatialTransfomerFixedOffset_15719580304089
// MI455X (gfx1250) — compile-verified
//
#include <hip/hip_runtime.h>

// SpatialTransformerFixedOffset: bilinear warp with a batch-invariant offset grid.
//   feature: (4,64,320,576) f32, offset: (1,320,576,2) f32, out: (4,64,320,576) f32
// Memory-bound: ~379 MB of traffic -> ~16 us floor at 23.3 TB/s. No matmul
// structure, so no WMMA; the optimization targets are data movement:
//  - per-pixel tap indices/weights computed once, reused across 32 planes/block
//  - non-temporal output stores (keep feature resident in the 192 MB L2)
//  - gfx1250 global_prefetch_b8 ahead of the plane loop

#define N_TOT   4
#define C_CH    64
#define H_DIM   320
#define W_DIM   576
#define HW      (H_DIM * W_DIM)      // 184320 = 720 * 256
#define PLANES  (N_TOT * C_CH)       // 256
#define PPB     32                   // planes per block  -> grid.y = 8
#define TPB     256                  // 8 wave32 per block

__global__ __launch_bounds__(TPB)
void st_warp_kernel(const float* __restrict__ feature,
                    const float* __restrict__ offset,
                    float* __restrict__ out)
{
    const int p = blockIdx.x * TPB + threadIdx.x;      // pixel id, always < HW
    const int y = p / W_DIM;
    const int x = p - y * W_DIM;

    // --- per-pixel sampling parameters (shared by all 256 planes) ---
    const float2 off = reinterpret_cast<const float2*>(offset)[p];
    const float sx = (float)x + off.x;                 // absolute sample x
    const float sy = (float)y + off.y;                 // absolute sample y

    const float x0f = floorf(sx);
    const float y0f = floorf(sy);
    const float wx1 = sx - x0f, wx0 = 1.0f - wx1;
    const float wy1 = sy - y0f, wy0 = 1.0f - wy1;

    const int x0 = (int)x0f, y0 = (int)y0f;
    const int x1 = x0 + 1,   y1 = y0 + 1;

    const bool vx0 = (x0 >= 0) && (x0 < W_DIM);
    const bool vx1 = (x1 >= 0) && (x1 < W_DIM);
    const bool vy0 = (y0 >= 0) && (y0 < H_DIM);
    const bool vy1 = (y1 >= 0) && (y1 < H_DIM);

    const int cx0 = min(max(x0, 0), W_DIM - 1);
    const int cx1 = min(max(x1, 0), W_DIM - 1);
    const int cy0 = min(max(y0, 0), H_DIM - 1);
    const int cy1 = min(max(y1, 0), H_DIM - 1);

    // fold zeros-padding masks into the bilinear weights
    const float w00 = (vy0 && vx0) ? (wy0 * wx0) : 0.0f;
    const float w01 = (vy0 && vx1) ? (wy0 * wx1) : 0.0f;
    const float w10 = (vy1 && vx0) ? (wy1 * wx0) : 0.0f;
    const float w11 = (vy1 && vx1) ? (wy1 * wx1) : 0.0f;

    const int i00 = cy0 * W_DIM + cx0;
    const int i01 = cy0 * W_DIM + cx1;
    const int i10 = cy1 * W_DIM + cx0;
    const int i11 = cy1 * W_DIM + cx1;

    // --- plane loop: 4 gathers + weighted sum + NT store per plane ---
    const int plane0 = blockIdx.y * PPB;
    const float* __restrict__ fbase = feature + (size_t)plane0 * HW;
    float* __restrict__ obase       = out     + (size_t)plane0 * HW + p;

#pragma unroll 8
    for (int j = 0; j < PPB; ++j) {
        if ((j & 7) == 0) {
            // gfx1250 global_prefetch_b8: pull the v00 line of the plane
            // 8 iterations ahead into cache (speculative, counter-free)
            const int pf = plane0 + j + 8;
            if (pf < PLANES) {
                __builtin_prefetch(fbase + (size_t)(j + 8) * HW + i00, 0, 1);
            }
        }
        const float* __restrict__ f = fbase + (size_t)j * HW;
        const float v00 = f[i00];
        const float v01 = f[i01];
        const float v10 = f[i10];
        const float v11 = f[i11];
        const float r = v00 * w00 + v01 * w01 + v10 * w10 + v11 * w11;
        // never re-read: non-temporal store protects feature lines in L2
        __builtin_nontemporal_store(r, obase + (size_t)j * HW);
    }
}

extern "C" void kernel_launch(void* const* d_in, const int* in_sizes, int n_in,
                              void* d_out, int out_size, void* d_ws, size_t ws_size,
                              hipStream_t stream)
{
    const float* feature = (const float*)d_in[0];  // (4,64,320,576) f32
    const float* offset  = (const float*)d_in[1];  // (1,320,576,2)  f32
    float* out = (float*)d_out;                    // (4,64,320,576) f32

    dim3 grid(HW / TPB, PLANES / PPB);             // (720, 8)
    st_warp_kernel<<<grid, TPB, 0, stream>>>(feature, offset, out);
}